// SentClassifier_38139309588968
// MI455X (gfx1250) — compile-verified
//
#include <hip/hip_runtime.h>
#include <math.h>

#define NB 4096   // batch
#define NC 1024   // feature / class dim
#define TPG 4     // column tiles processed simultaneously per wave

typedef __attribute__((ext_vector_type(2))) float v2f;
typedef __attribute__((ext_vector_type(8))) float v8f;

// ---------------------------------------------------------------- init
__global__ void init_kernel(float* acc) {
    acc[0] = 0.0f;   // ce sum
    acc[1] = 0.0f;   // info-nce sum
}

// ---------------------------------------------------------------- row stats
__global__ __launch_bounds__(256)
void row_stats_kernel(const float* __restrict__ outs,
                      const float* __restrict__ centers,
                      const int*   __restrict__ labels,
                      float* __restrict__ sq,
                      float* __restrict__ pos_sim,
                      float* __restrict__ acc) {
    const int row = blockIdx.x;
    const int t   = threadIdx.x;
    const float* x   = outs + (size_t)row * NC;
    const int    lab = labels[row];
    const float* cen = centers + (size_t)lab * NC;

    float mymax = -3.0e38f, mysq = 0.f, mypos = 0.f;
    for (int c = t; c < NC; c += 256) {
        float v = x[c];
        mymax = fmaxf(mymax, v);
        mysq  = fmaf(v, v, mysq);
        mypos = fmaf(v, cen[c], mypos);
    }
    __shared__ float smax[256], s1[256], s2[256];
    smax[t] = mymax; s1[t] = mysq; s2[t] = mypos;
    __syncthreads();
    for (int off = 128; off > 0; off >>= 1) {
        if (t < off) {
            smax[t] = fmaxf(smax[t], smax[t + off]);
            s1[t] += s1[t + off];
            s2[t] += s2[t + off];
        }
        __syncthreads();
    }
    const float rmax    = smax[0];
    const float tot_sq  = s1[0];
    const float tot_pos = s2[0];
    __syncthreads();

    float myse = 0.f;
    for (int c = t; c < NC; c += 256) myse += __expf(x[c] - rmax);
    s1[t] = myse;
    __syncthreads();
    for (int off = 128; off > 0; off >>= 1) {
        if (t < off) s1[t] += s1[t + off];
        __syncthreads();
    }
    if (t == 0) {
        sq[row]      = tot_sq;
        pos_sim[row] = tot_pos;
        float xl = x[lab];
        float ce = -(xl - rmax - logf(s1[0]));   // -logp[label]
        atomicAdd(&acc[0], ce);
    }
}

// ---------------------------------------------------------------- gram + hard argmax
__global__ __launch_bounds__(128)
void gram_hard_kernel(const float* __restrict__ X,
                      const int*   __restrict__ labels,
                      const float* __restrict__ sq,
                      int* __restrict__ hard_idx) {
    const int row0 = blockIdx.x * 16;
    const int lane = threadIdx.x & 31;
    const int wave = threadIdx.x >> 5;
    const int l15  = lane & 15;
    const int hi   = lane >> 4;       // 0: rows 0..7, 1: rows 8..15 of tile
    const int kh   = hi * 2;          // K sub-offset per ISA f32 A/B layout

    int rl[8];
#pragma unroll
    for (int v = 0; v < 8; ++v) rl[v] = labels[row0 + hi * 8 + v];

    const float* arow = X + (size_t)(row0 + l15) * NC + kh;

    float bv[8]; int bi[8];
#pragma unroll
    for (int v = 0; v < 8; ++v) { bv[v] = -3.0e38f; bi[v] = 0x7fffffff; }

    // 256 column tiles total; wave takes groups of TPG consecutive tiles.
    for (int tg = wave * TPG; tg < NB / 16; tg += 4 * TPG) {
        const float* brow[TPG];
        int jj[TPG]; float sqj[TPG]; int lj[TPG];
#pragma unroll
        for (int u = 0; u < TPG; ++u) {
            int j   = (tg + u) * 16 + l15;
            jj[u]   = j;
            sqj[u]  = sq[j];
            lj[u]   = labels[j];
            brow[u] = X + (size_t)j * NC + kh;
        }

        v8f d[TPG];
        const v8f zero = {};
#pragma unroll
        for (int u = 0; u < TPG; ++u) d[u] = zero;

        // ---- software-pipelined K loop ----
        v2f a = *(const v2f*)(arow);
        v2f b[TPG];
#pragma unroll
        for (int u = 0; u < TPG; ++u) b[u] = *(const v2f*)(brow[u]);

#pragma unroll 2
        for (int k0 = 0; k0 < NC - 4; k0 += 4) {
            const int kn = k0 + 4;
            v2f an = *(const v2f*)(arow + kn);       // prefetch next step
            v2f bn[TPG];
#pragma unroll
            for (int u = 0; u < TPG; ++u) bn[u] = *(const v2f*)(brow[u] + kn);
#pragma unroll
            for (int u = 0; u < TPG; ++u)
                d[u] = __builtin_amdgcn_wmma_f32_16x16x4_f32(
                           false, a, false, b[u], (short)0, d[u], false, false);
            a = an;
#pragma unroll
            for (int u = 0; u < TPG; ++u) b[u] = bn[u];
        }
#pragma unroll
        for (int u = 0; u < TPG; ++u)                // pipeline tail
            d[u] = __builtin_amdgcn_wmma_f32_16x16x4_f32(
                       false, a, false, b[u], (short)0, d[u], false, false);

        // ---- running masked argmax (sim == sq[j] - 2*dot up to per-row consts)
#pragma unroll
        for (int u = 0; u < TPG; ++u) {
#pragma unroll
            for (int v = 0; v < 8; ++v) {
                float val = sqj[u] - 2.0f * d[u][v];
                bool  upd = (lj[u] != rl[v]) &&
                            (val > bv[v] || (val == bv[v] && jj[u] < bi[v]));
                bv[v] = upd ? val   : bv[v];
                bi[v] = upd ? jj[u] : bi[v];
            }
        }
    }

    // reduce across the 16 lanes of each half (same row set per half)
#pragma unroll
    for (int m = 1; m < 16; m <<= 1) {
#pragma unroll
        for (int v = 0; v < 8; ++v) {
            float ov = __shfl_xor(bv[v], m, 32);
            int   oi = __shfl_xor(bi[v], m, 32);
            bool upd = (ov > bv[v]) || (ov == bv[v] && oi < bi[v]);
            bv[v] = upd ? ov : bv[v];
            bi[v] = upd ? oi : bi[v];
        }
    }

    __shared__ float sval[4][16];
    __shared__ int   sidx[4][16];
    if (l15 == 0) {
#pragma unroll
        for (int v = 0; v < 8; ++v) {
            sval[wave][hi * 8 + v] = bv[v];
            sidx[wave][hi * 8 + v] = bi[v];
        }
    }
    __syncthreads();
    if (threadIdx.x < 16) {
        float best  = sval[0][threadIdx.x];
        int   besti = sidx[0][threadIdx.x];
        for (int w = 1; w < 4; ++w) {
            float ov = sval[w][threadIdx.x];
            int   oi = sidx[w][threadIdx.x];
            bool upd = (ov > best) || (ov == best && oi < besti);
            best  = upd ? ov : best;
            besti = upd ? oi : besti;
        }
        hard_idx[row0 + threadIdx.x] = besti;
    }
}

// ---------------------------------------------------------------- negatives + info-nce
__device__ __forceinline__ unsigned int hash_u32(unsigned int a) {
    a ^= a >> 16; a *= 0x7feb352dU;
    a ^= a >> 15; a *= 0x846ca68bU;
    a ^= a >> 16;
    return a;
}

__global__ __launch_bounds__(256)
void negatives_kernel(const float* __restrict__ outs,
                      const int*   __restrict__ labels,
                      const float* __restrict__ pos_sim,
                      const int*   __restrict__ hard_idx,
                      float* __restrict__ acc) {
    const int i  = blockIdx.x;
    const int t  = threadIdx.x;
    const int li = labels[i];

    unsigned int bestr = 0; int bestj = -1;
    for (int j = t; j < NB; j += 256) {
        if (labels[j] != li) {
            unsigned int h = hash_u32((unsigned int)(i * NB + j) ^ 0x9e3779b9U);
            bool upd = (bestj < 0) || (h > bestr) || (h == bestr && j < bestj);
            bestr = upd ? h : bestr;
            bestj = upd ? j : bestj;
        }
    }
    __shared__ unsigned int sr[256];
    __shared__ int          sj[256];
    sr[t] = bestr; sj[t] = bestj;
    __syncthreads();
    for (int off = 128; off > 0; off >>= 1) {
        if (t < off) {
            bool take = (sj[t] < 0) ||
                        (sj[t + off] >= 0 &&
                         (sr[t + off] > sr[t] ||
                          (sr[t + off] == sr[t] && sj[t + off] < sj[t])));
            if (take) { sr[t] = sr[t + off]; sj[t] = sj[t + off]; }
        }
        __syncthreads();
    }
    int rj = sj[0];
    if (rj < 0) rj = (i + 1) & (NB - 1);
    const int hj = hard_idx[i];

    const float* xi = outs + (size_t)i  * NC;
    const float* xr = outs + (size_t)rj * NC;
    const float* xh = outs + (size_t)hj * NC;
    float dr = 0.f, dh = 0.f;
    for (int c = t; c < NC; c += 256) {
        float v = xi[c];
        dr = fmaf(v, xr[c], dr);
        dh = fmaf(v, xh[c], dh);
    }
    __shared__ float sdr[256], sdh[256];
    sdr[t] = dr; sdh[t] = dh;
    __syncthreads();
    for (int off = 128; off > 0; off >>= 1) {
        if (t < off) { sdr[t] += sdr[t + off]; sdh[t] += sdh[t + off]; }
        __syncthreads();
    }
    if (t == 0) {
        float p = pos_sim[i], r = sdr[0], h = sdh[0];
        float m = fmaxf(p, fmaxf(r, h));
        float lse = m + logf(__expf(p - m) + __expf(r - m) + __expf(h - m));
        atomicAdd(&acc[1], lse - p);   // -log_softmax[...,0]
    }
}

// ---------------------------------------------------------------- finalize
__global__ void finalize_kernel(const float* acc, float* out) {
    out[0] = acc[0] / (float)NB + 0.1f * (acc[1] / (float)NB);
}

// ---------------------------------------------------------------- launch
extern "C" void kernel_launch(void* const* d_in, const int* in_sizes, int n_in,
                              void* d_out, int out_size, void* d_ws, size_t ws_size,
                              hipStream_t stream) {
    const float* outs    = (const float*)d_in[0];   // [4096,1024]
    const float* centers = (const float*)d_in[1];   // [1024,1024]
    const int*   labels  = (const int*)  d_in[2];   // [4096]
    float* out = (float*)d_out;

    char* ws = (char*)d_ws;
    float* sq       = (float*)(ws);
    int*   hard_idx = (int*)  (ws + 1 * NB * 4);
    float* pos_sim  = (float*)(ws + 2 * NB * 4);
    float* acc      = (float*)(ws + 3 * NB * 4);    // [0]=ce sum, [1]=info sum

    init_kernel<<<1, 1, 0, stream>>>(acc);
    row_stats_kernel<<<NB, 256, 0, stream>>>(outs, centers, labels, sq, pos_sim, acc);
    gram_hard_kernel<<<NB / 16, 128, 0, stream>>>(outs, labels, sq, hard_idx);
    negatives_kernel<<<NB, 256, 0, stream>>>(outs, labels, pos_sim, hard_idx, acc);
    finalize_kernel<<<1, 1, 0, stream>>>(acc, out);
}